// SingleHeadGATConv_996432413193
// MI455X (gfx1250) — compile-verified
//
#include <hip/hip_runtime.h>
#include <hip/hip_bf16.h>
#include <math.h>

#define ALPHA 0.2f
#define IN_F 128
#define OUT_F 64

typedef __attribute__((ext_vector_type(2))) float v2f;
typedef __attribute__((ext_vector_type(8))) float v8f;

// Padded row stride (in float2 units) for the swizzled W tile in LDS.
// 80 float2 = 640 B => lane-half 1 (row u+1) starts at bank 32, giving a
// conflict-free 64-bank access pattern for the 8-byte B-fragment loads.
#define WSWZ_STRIDE 80

// ---------------------------------------------------------------------------
// Kernel 1: Wh = x @ W via V_WMMA_F32_16X16X4_F32.
// One wave32 per 16-row M-tile, covering all 4 N-tiles (N=64) with 4 f32
// accumulators, K=128 stepped by 4.
//
// W is staged in LDS in fragment-major float2 form:
//   Wswz[u][col] = { W[2u][col], W[2u+1][col] },  u in [0,64)
// so a B fragment (half-lane K split: lanes 0-15 -> K={k0,k0+1},
// lanes 16-31 -> K={k0+2,k0+3}) is a single aligned ds_load_b64:
//   u = k0/2 + (lane>>4)
// ---------------------------------------------------------------------------
__global__ void __launch_bounds__(256) gat_gemm_wh(const float* __restrict__ x,
                                                   const float* __restrict__ W,
                                                   float* __restrict__ Wh,
                                                   int nNodes) {
    __shared__ v2f Wswz[64 * WSWZ_STRIDE];  // 40 KB

    // Swizzled, coalesced fill: thread handles (u, col); reads W[2u][col],
    // W[2u+1][col] (lanes sweep col => contiguous 256B bursts per row).
    for (int idx = threadIdx.x; idx < 64 * 64; idx += 256) {
        const int u = idx >> 6;
        const int col = idx & 63;
        v2f w;
        w.x = W[(2 * u) * OUT_F + col];
        w.y = W[(2 * u + 1) * OUT_F + col];
        Wswz[u * WSWZ_STRIDE + col] = w;
    }
    __syncthreads();

    const int wave  = threadIdx.x >> 5;
    const int lane  = threadIdx.x & 31;
    const int mtile = blockIdx.x * 8 + wave;
    if (mtile * 16 >= nNodes) return;

    // A-fragment 16x4 f32 layout (ISA 7.12.2): lanes 0-15 -> M=lane, K={k,k+1};
    // lanes 16-31 -> M=lane-16, K={k+2,k+3}.
    const int mrow  = mtile * 16 + (lane & 15);
    const int khalf = (lane >> 4) * 2;  // 0 or 2
    const int col0  = lane & 15;        // N within tile (B/C/D row striped across lanes)
    const int uhalf = lane >> 4;        // B fragment row-pair select

    v8f acc0 = {}, acc1 = {}, acc2 = {}, acc3 = {};

    const float* xrow = x + (size_t)mrow * IN_F;

    for (int k0 = 0; k0 < IN_F; k0 += 4) {
        // A fragment: aligned 8-byte global load (kk is even).
        const int kk = k0 + khalf;
        const v2f a = *(const v2f*)(xrow + kk);

        // B fragments: one aligned ds_load_b64 each, no repacking movs.
        const v2f* wrow = &Wswz[((k0 >> 1) + uhalf) * WSWZ_STRIDE];
        const v2f b0 = wrow[col0];
        const v2f b1 = wrow[16 + col0];
        const v2f b2 = wrow[32 + col0];
        const v2f b3 = wrow[48 + col0];

        // 8-arg form: (neg_a, A, neg_b, B, c_mod, C, reuse_a, reuse_b)
        acc0 = __builtin_amdgcn_wmma_f32_16x16x4_f32(false, a, false, b0, (short)0, acc0, false, false);
        acc1 = __builtin_amdgcn_wmma_f32_16x16x4_f32(false, a, false, b1, (short)0, acc1, false, false);
        acc2 = __builtin_amdgcn_wmma_f32_16x16x4_f32(false, a, false, b2, (short)0, acc2, false, false);
        acc3 = __builtin_amdgcn_wmma_f32_16x16x4_f32(false, a, false, b3, (short)0, acc3, false, false);
    }

    // C/D layout: VGPR r -> M = r (lanes 0-15) / r+8 (lanes 16-31), N = lane%16.
    const int rbase = mtile * 16 + ((lane >> 4) ? 8 : 0);
    for (int r = 0; r < 8; ++r) {
        float* o = &Wh[(size_t)(rbase + r) * OUT_F + col0];
        o[0]  = acc0[r];
        o[16] = acc1[r];
        o[32] = acc2[r];
        o[48] = acc3[r];
    }
}

// ---------------------------------------------------------------------------
// Kernel 2: per-edge attention logits.
// One wave32 per edge; lane handles features {lane, lane+32}; butterfly
// reduce; leaky_relu. Wh (12.8 MB) is L2-resident so gathers stay on-chip.
// ---------------------------------------------------------------------------
__global__ void __launch_bounds__(256) gat_edge_scores(const float* __restrict__ Wh,
                                                       const int* __restrict__ ei,
                                                       const float* __restrict__ a,
                                                       float* __restrict__ e,
                                                       int nEdges) {
    const int g    = blockIdx.x * 256 + threadIdx.x;
    const int edge = g >> 5;
    const int lane = g & 31;
    if (edge >= nEdges) return;

    const int s = ei[edge];
    const int d = ei[nEdges + edge];
    const float* whs = Wh + (size_t)s * OUT_F;
    const float* whd = Wh + (size_t)d * OUT_F;

    float v = whs[lane]      * a[lane]
            + whs[lane + 32] * a[lane + 32]
            + whd[lane]      * a[OUT_F + lane]
            + whd[lane + 32] * a[OUT_F + lane + 32];

    for (int off = 16; off > 0; off >>= 1)
        v += __shfl_xor(v, off, 32);

    if (lane == 0)
        e[edge] = (v > 0.0f) ? v : (ALPHA * v);
}

// ---------------------------------------------------------------------------
// Kernel 3: global softmax stats (reference applies softmax over ALL edges).
// Single 1024-thread block: pass 1 = max, pass 2 = sum(exp(e - max)).
// ---------------------------------------------------------------------------
__global__ void gat_softmax_stats(const float* __restrict__ e, int n,
                                  float* __restrict__ scal) {
    __shared__ float sdata[1024];
    const int tid = threadIdx.x;

    float m = -INFINITY;
    for (int i = tid; i < n; i += 1024) m = fmaxf(m, e[i]);
    sdata[tid] = m;
    __syncthreads();
    for (int s = 512; s > 0; s >>= 1) {
        if (tid < s) sdata[tid] = fmaxf(sdata[tid], sdata[tid + s]);
        __syncthreads();
    }
    const float gmax = sdata[0];
    __syncthreads();

    float sum = 0.0f;
    for (int i = tid; i < n; i += 1024) sum += __expf(e[i] - gmax);
    sdata[tid] = sum;
    __syncthreads();
    for (int s = 512; s > 0; s >>= 1) {
        if (tid < s) sdata[tid] += sdata[tid + s];
        __syncthreads();
    }
    if (tid == 0) {
        scal[0] = gmax;
        scal[1] = sdata[0];
    }
}

// ---------------------------------------------------------------------------
// Kernel 4: zero the segment-sum accumulator (replay-safe).
// ---------------------------------------------------------------------------
__global__ void gat_zero(float* __restrict__ p, int n) {
    const int i = blockIdx.x * blockDim.x + threadIdx.x;
    if (i < n) p[i] = 0.0f;
}

// ---------------------------------------------------------------------------
// Kernel 5: scatter msgs = Wh_dst * attention into acc[src] via f32 atomics
// (GLOBAL_ATOMIC_ADD_F32; these resolve in L2 since acc is 12.8 MB).
// ---------------------------------------------------------------------------
__global__ void __launch_bounds__(256) gat_edge_scatter(const float* __restrict__ Wh,
                                                        const int* __restrict__ ei,
                                                        const float* __restrict__ e,
                                                        const float* __restrict__ scal,
                                                        float* __restrict__ acc,
                                                        int nEdges) {
    const int g    = blockIdx.x * 256 + threadIdx.x;
    const int edge = g >> 5;
    const int lane = g & 31;
    if (edge >= nEdges) return;

    const float att = __expf(e[edge] - scal[0]) / scal[1];
    const int s = ei[edge];
    const int d = ei[nEdges + edge];
    const float* whd = Wh + (size_t)d * OUT_F;
    float* o = acc + (size_t)s * OUT_F;

    atomicAdd(&o[lane],      whd[lane]      * att);
    atomicAdd(&o[lane + 32], whd[lane + 32] * att);
}

// ---------------------------------------------------------------------------
// Kernel 6: ELU epilogue -> d_out (written exactly once per launch).
// ---------------------------------------------------------------------------
__global__ void gat_elu(const float* __restrict__ acc, float* __restrict__ out, int n) {
    const int i = blockIdx.x * blockDim.x + threadIdx.x;
    if (i < n) {
        const float v = acc[i];
        out[i] = (v > 0.0f) ? v : (__expf(v) - 1.0f);
    }
}

extern "C" void kernel_launch(void* const* d_in, const int* in_sizes, int n_in,
                              void* d_out, int out_size, void* d_ws, size_t ws_size,
                              hipStream_t stream) {
    const float* x  = (const float*)d_in[0];   // (nNodes, 128) f32
    const int*   ei = (const int*)d_in[1];     // (2, nEdges) int32
    const float* W  = (const float*)d_in[2];   // (128, 64) f32
    const float* a  = (const float*)d_in[3];   // (128, 1) f32

    const int nNodes = in_sizes[0] / IN_F;
    const int nEdges = in_sizes[1] / 2;

    // Workspace layout (floats): Wh | e | acc | {max,sum}
    float* ws   = (float*)d_ws;
    float* Wh   = ws;
    float* e    = Wh + (size_t)nNodes * OUT_F;
    float* acc  = e + nEdges;
    float* scal = acc + (size_t)nNodes * OUT_F;

    const int mtiles = (nNodes + 15) / 16;
    gat_gemm_wh<<<(mtiles + 7) / 8, 256, 0, stream>>>(x, W, Wh, nNodes);

    const int edgeThreads = nEdges * 32;
    gat_edge_scores<<<(edgeThreads + 255) / 256, 256, 0, stream>>>(Wh, ei, a, e, nEdges);

    gat_softmax_stats<<<1, 1024, 0, stream>>>(e, nEdges, scal);

    const int accN = nNodes * OUT_F;
    gat_zero<<<(accN + 255) / 256, 256, 0, stream>>>(acc, accN);

    gat_edge_scatter<<<(edgeThreads + 255) / 256, 256, 0, stream>>>(Wh, ei, e, scal, acc, nEdges);

    gat_elu<<<(accN + 255) / 256, 256, 0, stream>>>(acc, (float*)d_out, accN);
}